// Loss_31078383354739
// MI455X (gfx1250) — compile-verified
//
#include <hip/hip_runtime.h>
#include <hip/hip_bf16.h>

// DETR matching cost: cost[b,n,m] = L1(bbox) + GIoU-style loss - gathered class prob.
// Class term computed as an EXACT one-hot fp32 matmul via V_WMMA_F32_16X16X4_F32.

typedef __attribute__((ext_vector_type(2))) float v2f;
typedef __attribute__((ext_vector_type(8))) float v8f;

#define B_DIM 64
#define N_DIM 900
#define M_DIM 300
#define NCLS  92   // NUM_CLASSES + 1

__global__ __launch_bounds__(256) void detr_cost_wmma(
    const float* __restrict__ bbox_pred,    // [B, N, 4]
    const float* __restrict__ labels_pred,  // [B, N, 92]
    const float* __restrict__ bbox_gt,      // [B, M, 4]
    const int*   __restrict__ labels_gt,    // [B, M]
    float*       __restrict__ out)          // [B, N, M]
{
    const int tid  = threadIdx.x;
    const int wave = tid >> 5;          // 8 waves per block (wave32)
    const int lane = tid & 31;
    const int lo   = lane & 15;
    const int hi   = lane >> 4;

    const int b      = blockIdx.z;
    const int n_base = blockIdx.y * 128 + wave * 16;   // 16 n-rows per wave
    const int m_base = blockIdx.x * 16;                // 16 m-cols per block

    // ---- stage per-n (pred) derived values in LDS, one 16-row strip per wave ----
    __shared__ float s_px[8][16], s_py[8][16], s_pw[8][16], s_ph[8][16];
    __shared__ float s_pux[8][16], s_puy[8][16], s_pdx[8][16], s_pdy[8][16];
    __shared__ float s_parea[8][16];

    if (lane < 16) {
        int n = n_base + lane; if (n > N_DIM - 1) n = N_DIM - 1;   // clamp tail
        const float4 pb = *(const float4*)(bbox_pred + ((size_t)b * N_DIM + n) * 4);
        s_px[wave][lane] = pb.x;  s_py[wave][lane] = pb.y;
        s_pw[wave][lane] = pb.z;  s_ph[wave][lane] = pb.w;
        s_pux[wave][lane] = pb.x - 0.5f * pb.z;
        s_puy[wave][lane] = pb.y - 0.5f * pb.w;
        s_pdx[wave][lane] = pb.x + 0.5f * pb.z;
        s_pdy[wave][lane] = pb.y + 0.5f * pb.w;
        s_parea[wave][lane] = fmaxf(pb.z + 1.f, 0.f) * fmaxf(pb.w + 1.f, 0.f);
    }
    __syncthreads();

    // ---- per-lane gt-side values (m fixed per lane; lanes 16..31 mirror 0..15) ----
    const int m  = m_base + lo;
    int mc = m; if (mc > M_DIM - 1) mc = M_DIM - 1;
    const float4 gb = *(const float4*)(bbox_gt + ((size_t)b * M_DIM + mc) * 4);
    const int   lg = labels_gt[(size_t)b * M_DIM + mc];
    const float gux = gb.x - 0.5f * gb.z, guy = gb.y - 0.5f * gb.w;
    const float gdx = gb.x + 0.5f * gb.z, gdy = gb.y + 0.5f * gb.w;
    const float garea = fmaxf(gb.z + 1.f, 0.f) * fmaxf(gb.w + 1.f, 0.f);

    // ---- class term: acc(16x16) = labels_pred_tile(16x92) x onehot(92x16) ----
    // A (16x4 f32): lane holds row = lane&15; VGPR0 = K(+2*hi), VGPR1 = K+1(+2*hi)
    int arow = n_base + lo; if (arow > N_DIM - 1) arow = N_DIM - 1;
    const float* ap = labels_pred + ((size_t)b * N_DIM + arow) * NCLS + 2 * hi;

    v8f acc = {};
    #pragma unroll
    for (int k0 = 0; k0 < NCLS; k0 += 4) {
        v2f a = *(const v2f*)(ap + k0);                     // 8B aligned
        v2f bf;                                             // one-hot B fragment
        bf.x = (lg == (k0 + 2 * hi))     ? 1.f : 0.f;       // row K = k0 + 2*hi
        bf.y = (lg == (k0 + 2 * hi + 1)) ? 1.f : 0.f;       // row K = k0 + 1 + 2*hi
        // 8 args: (neg_a, A, neg_b, B, c_mod, C, reuse_a, reuse_b)
        acc = __builtin_amdgcn_wmma_f32_16x16x4_f32(
            false, a, false, bf, (short)0, acc, false, false);
    }

    // ---- epilogue: per-pair L1 + GIoU terms; D layout: VGPR r -> n = r + 8*hi ----
    #pragma unroll
    for (int r = 0; r < 8; ++r) {
        const int nl = r + 8 * hi;
        const int n  = n_base + nl;

        const float bl = fabsf(s_px[wave][nl] - gb.x) + fabsf(s_py[wave][nl] - gb.y)
                       + fabsf(s_pw[wave][nl] - gb.z) + fabsf(s_ph[wave][nl] - gb.w);

        const float pux = s_pux[wave][nl], puy = s_puy[wave][nl];
        const float pdx = s_pdx[wave][nl], pdy = s_pdy[wave][nl];

        const float iw = fmaxf(fminf(pdx, gdx) - fmaxf(pux, gux) + 1.f, 0.f);
        const float ih = fmaxf(fminf(pdy, gdy) - fmaxf(puy, guy) + 1.f, 0.f);
        const float inter = iw * ih;
        const float uni   = s_parea[wave][nl] + garea - inter;

        const float bw = fmaxf(fmaxf(pdx, gdx) - fminf(pux, gux) + 1.f, 0.f);
        const float bh = fmaxf(fmaxf(pdy, gdy) - fminf(puy, guy) + 1.f, 0.f);
        const float bound = bw * bh;

        // cost = L1 - inter/union + (bound-union)/bound - prob
        const float cost = bl + 1.f - inter / uni - uni / bound - acc[r];

        if (n < N_DIM && m < M_DIM)
            out[((size_t)b * N_DIM + n) * M_DIM + m] = cost;
    }
}

extern "C" void kernel_launch(void* const* d_in, const int* in_sizes, int n_in,
                              void* d_out, int out_size, void* d_ws, size_t ws_size,
                              hipStream_t stream) {
    (void)in_sizes; (void)n_in; (void)out_size; (void)d_ws; (void)ws_size;
    const float* bbox_pred   = (const float*)d_in[0];
    const float* labels_pred = (const float*)d_in[1];
    const float* bbox_gt     = (const float*)d_in[2];
    const int*   labels_gt   = (const int*)d_in[3];
    float* out = (float*)d_out;

    dim3 grid((M_DIM + 15) / 16,      // 19 m-tiles of 16
              (N_DIM + 127) / 128,    // 8 n-tiles of 128 (16 per wave)
              B_DIM);                 // 64 batches
    detr_cost_wmma<<<grid, 256, 0, stream>>>(bbox_pred, labels_pred, bbox_gt,
                                             labels_gt, out);
}